// MyLayer_67946382623286
// MI455X (gfx1250) — compile-verified
//
#include <hip/hip_runtime.h>
#include <hip/hip_bf16.h>

#define NODES   10000
#define BATCH   8
#define FIN     128
#define NROWS   (NODES * BATCH)   // 80000 GEMM rows (r = b*NODES + m)

typedef __attribute__((ext_vector_type(2))) float v2f;
typedef __attribute__((ext_vector_type(8))) float v8f;

// ---------------------------------------------------------------------------
// Projection: p_k[m*8+b] = sum_fin x[b,m,fin] * w1[fin*3 + k],  k = 0,1,2
// One wave per 16-row tile, V_WMMA_F32_16X16X4_F32 over K=128 (32 steps).
// A[mm][kk] = x[row0+mm][k0+kk]; B[kk][nn] = w1[(k0+kk)*3+nn] (nn<3 else 0).
// B-fragments are tile-invariant -> staged once per block in LDS (branchless).
// ---------------------------------------------------------------------------
__global__ __launch_bounds__(256)
void proj_wmma_kernel(const float* __restrict__ x, const float* __restrict__ w1,
                      float* __restrict__ p /* 3 buffers of NROWS floats */) {
    __shared__ float bfr[32 * 32 * 2];               // [step][lane][2] = 8 KB

    // Branchless fill: clamp index, mask value. One pass, 4 entries/thread.
    for (int i = threadIdx.x; i < 1024; i += 256) {
        const int s  = i >> 5;
        const int ln = i & 31;
        const int lr = ln & 15;
        const int kh = ln >> 4;
        const int k  = 4 * s + 2 * kh;
        const int   nn  = (lr < 3) ? lr  : 0;
        const float msk = (lr < 3) ? 1.f : 0.f;
        bfr[i * 2]     = w1[k * 3 + nn]       * msk;
        bfr[i * 2 + 1] = w1[(k + 1) * 3 + nn] * msk;
    }
    __syncthreads();

    const int wave = threadIdx.x >> 5;
    const int lane = threadIdx.x & 31;
    const int tile = blockIdx.x * 8 + wave;          // 0..4999
    const int row0 = tile * 16;                      // tile never straddles b (10000 % 16 == 0)

    const int lrow  = lane & 15;                     // M index (A) / N index (B,D)
    const int khalf = lane >> 4;                     // 0: K={0,1}, 1: K={2,3}

    const float* xrow = x + (size_t)(row0 + lrow) * FIN + 2 * khalf;

    v8f c = {0.f, 0.f, 0.f, 0.f, 0.f, 0.f, 0.f, 0.f};

    #pragma unroll 8
    for (int s = 0; s < 32; ++s) {
        v2f a;
        a[0] = xrow[4 * s];
        a[1] = xrow[4 * s + 1];

        const v2f bf = *(const v2f*)&bfr[(s * 32 + lane) * 2];

        c = __builtin_amdgcn_wmma_f32_16x16x4_f32(
                /*neg_a=*/false, a, /*neg_b=*/false, bf,
                /*c_mod=*/(short)0, c, /*reuse_a=*/false, /*reuse_b=*/false);
    }

    // D: vgpr j, lanes 0-15 -> D[M=j][N=lane]; lanes 16-31 -> D[M=8+j][N=lane-16]
    if (lrow < 3) {
        float* dst = p + (size_t)lrow * NROWS;       // p0 / p1 / p2
        const int mbase = khalf * 8;
        #pragma unroll
        for (int j = 0; j < 8; ++j) {
            const int r = row0 + mbase + j;          // r = b*NODES + m
            const int b = r / NODES;
            const int m = r - b * NODES;
            dst[m * BATCH + b] = c[j];
        }
    }
}

// ---------------------------------------------------------------------------
// SpMM on [NODES, 8]: out[rows[e]*8+b] += vals[e] * in[cols[e]*8+b]
// ---------------------------------------------------------------------------
__global__ __launch_bounds__(256)
void spmm8_kernel(const int* __restrict__ rows, const int* __restrict__ cols,
                  const float* __restrict__ vals, const float* __restrict__ in,
                  float* __restrict__ out, int E) {
    const int t = blockIdx.x * blockDim.x + threadIdx.x;
    if (t >= E * BATCH) return;
    const int e = t >> 3;
    const int f = t & 7;
    const float v = vals[e];
    const int   c = cols[e];
    const int   r = rows[e];
    unsafeAtomicAdd(out + r * BATCH + f, v * in[c * BATCH + f]);
}

__global__ __launch_bounds__(256)
void zero_kernel(float* __restrict__ p, int n) {
    const int t = blockIdx.x * blockDim.x + threadIdx.x;
    if (t < n) p[t] = 0.f;
}

// h1 = relu(p0 + q1 + 2*q2 - p2 + bias)          (layer-1 output, [m*8+b])
__global__ __launch_bounds__(256)
void combine1_kernel(const float* __restrict__ p0, const float* __restrict__ q1,
                     const float* __restrict__ q2, const float* __restrict__ p2,
                     const float* __restrict__ bias, float* __restrict__ h1, int n) {
    const int t = blockIdx.x * blockDim.x + threadIdx.x;
    if (t >= n) return;
    float v = p0[t] + q1[t] + 2.f * q2[t] - p2[t] + bias[0];
    h1[t] = v > 0.f ? v : 0.f;
}

// out[b*M+m] = relu(w2[0]*y0 + w2[1]*y1 + w2[2]*(2*y2t - y0) + bias)
__global__ __launch_bounds__(256)
void combine2_kernel(const float* __restrict__ h1, const float* __restrict__ y1,
                     const float* __restrict__ y2t, const float* __restrict__ w2,
                     const float* __restrict__ bias, float* __restrict__ out, int n) {
    const int t = blockIdx.x * blockDim.x + threadIdx.x;
    if (t >= n) return;
    const float y0 = h1[t];
    const float Y1 = y1[t];
    const float Y2 = 2.f * y2t[t] - y0;
    float v = w2[0] * y0 + w2[1] * Y1 + w2[2] * Y2 + bias[0];
    v = v > 0.f ? v : 0.f;
    const int m = t >> 3;
    const int b = t & 7;
    out[b * NODES + m] = v;
}

// ---------------------------------------------------------------------------
extern "C" void kernel_launch(void* const* d_in, const int* in_sizes, int n_in,
                              void* d_out, int out_size, void* d_ws, size_t ws_size,
                              hipStream_t stream) {
    const float* x    = (const float*)d_in[0];
    const int*   rows = (const int*)  d_in[1];
    const int*   cols = (const int*)  d_in[2];
    const float* vals = (const float*)d_in[3];
    const float* w1   = (const float*)d_in[4];
    const float* w2   = (const float*)d_in[5];
    const float* bias = (const float*)d_in[6];
    float*       out  = (float*)d_out;
    const int E = in_sizes[1];

    float* ws  = (float*)d_ws;
    float* p0  = ws;                 // NROWS each
    float* p1  = ws + 1 * NROWS;
    float* p2  = ws + 2 * NROWS;
    float* h1  = ws + 3 * NROWS;
    float* q1  = ws + 4 * NROWS;     // L p1        (atomic dst)
    float* r_  = ws + 5 * NROWS;     // L p2        (atomic dst)
    float* q2  = ws + 6 * NROWS;     // L (L p2)    (atomic dst)
    float* y1  = ws + 7 * NROWS;     // L h1        (atomic dst)
    float* y2t = ws + 8 * NROWS;     // L y1        (atomic dst)

    const int ZN = 5 * NROWS;        // zero all 5 atomic accumulators
    zero_kernel<<<(ZN + 255) / 256, 256, 0, stream>>>(q1, ZN);

    // dense projection (WMMA): p0,p1,p2
    proj_wmma_kernel<<<(NROWS / 16) / 8, 256, 0, stream>>>(x, w1, p0);

    const int SG = (E * BATCH + 255) / 256;
    spmm8_kernel<<<SG, 256, 0, stream>>>(rows, cols, vals, p1, q1, E);   // q1 = L p1
    spmm8_kernel<<<SG, 256, 0, stream>>>(rows, cols, vals, p2, r_, E);   // r  = L p2
    spmm8_kernel<<<SG, 256, 0, stream>>>(rows, cols, vals, r_, q2, E);   // q2 = L r

    combine1_kernel<<<(NROWS + 255) / 256, 256, 0, stream>>>(p0, q1, q2, p2, bias, h1, NROWS);

    spmm8_kernel<<<SG, 256, 0, stream>>>(rows, cols, vals, h1, y1, E);   // y1  = L h1
    spmm8_kernel<<<SG, 256, 0, stream>>>(rows, cols, vals, y1, y2t, E);  // y2t = L y1

    combine2_kernel<<<(NROWS + 255) / 256, 256, 0, stream>>>(h1, y1, y2t, w2, bias, out, NROWS);
}